// IMUEKF_67070209294568
// MI455X (gfx1250) — compile-verified
//
#include <hip/hip_runtime.h>

typedef float v2f __attribute__((ext_vector_type(2)));
typedef float v8f __attribute__((ext_vector_type(8)));

#define LS 17          // LDS tile row stride (pad to avoid bank conflicts)
#define TILE (16 * LS) // 272 floats per padded 16x16 tile
#define STGF 1664      // staged floats per wave (6*272=1632, padded to 13*128)

static __device__ __forceinline__ int lane_id() { return (int)(threadIdx.x & 31u); }

// Cross-lane LDS handoff: LDS ops from one wave are processed IN ORDER by the
// DS pipe (ISA 7.3), so no hardware wait is required for store->load through
// LDS. We only need a compiler barrier so clang cannot reorder DS ops that
// alias cross-lane (it can only prove lane-local disjointness).
static __device__ __forceinline__ void lds_fence() {
  asm volatile("" ::: "memory");
}

// Full 16x16x16 f32 product: chain of 4 x V_WMMA_F32_16X16X4_F32.
static __device__ __forceinline__ v8f mm16(const float fa[8], const float fb[8], v8f c) {
#pragma unroll
  for (int s = 0; s < 4; ++s) {
    v2f a, b;
    a.x = fa[2 * s]; a.y = fa[2 * s + 1];
    b.x = fb[2 * s]; b.y = fb[2 * s + 1];
    c = __builtin_amdgcn_wmma_f32_16x16x4_f32(false, a, false, b, (short)0, c,
                                              false, false);
  }
  return c;
}

static __device__ __forceinline__ int kidx(int i, int h) {
  return ((i >> 1) << 2) + 2 * h + (i & 1);
}

// A-operand fragment from a zero-padded LDS tile. Also serves as the
// B-operand fragment of the tile's transpose (identical lane mapping).
static __device__ __forceinline__ void loadA_lds(const float* t, float f[8]) {
  int lane = lane_id(); int r = lane & 15; int h = lane >> 4;
#pragma unroll
  for (int i = 0; i < 8; ++i) f[i] = t[r * LS + kidx(i, h)];
}

// B-operand fragment from a zero-padded LDS tile.
static __device__ __forceinline__ void loadB_lds(const float* t, float f[8]) {
  int lane = lane_id(); int r = lane & 15; int h = lane >> 4;
#pragma unroll
  for (int i = 0; i < 8; ++i) f[i] = t[kidx(i, h) * LS + r];
}

// D/C layout: VGPR v, lanes 0-15 -> row v, lanes 16-31 -> row v+8; col = lane&15
static __device__ __forceinline__ void storeD(float* t, v8f d) {
  int lane = lane_id(); int c = lane & 15; int h = lane >> 4;
#pragma unroll
  for (int v = 0; v < 8; ++v) t[(v + 8 * h) * LS + c] = d[v];
}

// Zero the whole staged region with vector LDS stores (pre-pads the tiles).
static __device__ __forceinline__ void zero_stage(float* s) {
  int lane = lane_id();
  float4 z4 = make_float4(0.f, 0.f, 0.f, 0.f);
  float4* s4 = (float4*)s;
#pragma unroll
  for (int base = 0; base < STGF / 4; base += 32) s4[base + lane] = z4;
}

// Coalesced cooperative copy of a ROWSxCOLS row-major matrix into a padded tile.
template <int ROWS, int COLS>
static __device__ __forceinline__ void stage_mat(const float* __restrict__ src,
                                                 float* t) {
  int lane = lane_id();
#pragma unroll
  for (int base = 0; base < ROWS * COLS; base += 32) {
    int i = base + lane;
    if (i < ROWS * COLS) {
      int r = i / COLS, c = i - r * COLS;
      t[r * LS + c] = src[i];
    }
  }
}

__global__ __launch_bounds__(128) void ekf_wmma_kernel(
    const float* __restrict__ state, const float* __restrict__ obs,
    const float* __restrict__ u, const float* __restrict__ Pmat,
    const float* __restrict__ A, const float* __restrict__ Bm,
    const float* __restrict__ C, const float* __restrict__ D,
    const float* __restrict__ c1, const float* __restrict__ c2,
    const float* __restrict__ Q, const float* __restrict__ R,
    float* __restrict__ out_x, float* __restrict__ out_P, int BN) {
  __shared__ __align__(16) float stg[4][STGF];  // staged A,P,B,Q,C,R
  __shared__ float tiles[4][2][TILE];           // scratch t0, t3
  __shared__ float ev[4][8];

  int w = (int)(threadIdx.x >> 5);
  int lane = lane_id();
  int b = (int)blockIdx.x * 4 + w;
  if (b >= BN) return;  // wave-uniform guard; no block barriers used

  float* tA = stg[w] + 0 * TILE;
  float* tP = stg[w] + 1 * TILE;
  float* tB = stg[w] + 2 * TILE;
  float* tQ = stg[w] + 3 * TILE;
  float* tC = stg[w] + 4 * TILE;
  float* tR = stg[w] + 5 * TILE;
  float* t0 = tiles[w][0];
  float* t3 = tiles[w][1];
  float* t1 = tP;  // tP dead after step 1 -> reuse as scratch (occupancy)
  float* t2 = tQ;  // tQ dead after step 2 -> reuse as scratch (holds S, then K)

  const float* Ab  = A    + (size_t)b * 225;
  const float* Pb  = Pmat + (size_t)b * 225;
  const float* Bb  = Bm   + (size_t)b * 90;
  const float* Cb  = C    + (size_t)b * 90;
  const float* Db  = D    + (size_t)b * 36;
  const float* Qb  = Q    + (size_t)b * 36;
  const float* Rb  = R    + (size_t)b * 36;
  const float* xb  = state + (size_t)b * 15;
  const float* ub  = u    + (size_t)b * 6;
  const float* ob  = obs  + (size_t)b * 6;
  const float* c1b = c1   + (size_t)b * 15;
  const float* c2b = c2   + (size_t)b * 6;

  // ---- stage inputs into zero-padded LDS tiles (coalesced) ----
  zero_stage(stg[w]);
  stage_mat<15, 15>(Ab, tA);
  stage_mat<15, 15>(Pb, tP);
  stage_mat<15, 6>(Bb, tB);
  stage_mat<6, 6>(Qb, tQ);
  stage_mat<6, 15>(Cb, tC);
  stage_mat<6, 6>(Rb, tR);
  lds_fence();

  float fa[8], fb[8];
  v8f z = {0.f, 0.f, 0.f, 0.f, 0.f, 0.f, 0.f, 0.f};
  v8f acc;

  // step 1: t0 = A * P   (last use of tP as staged data)
  loadA_lds(tA, fa); loadB_lds(tP, fb);
  acc = mm16(fa, fb, z); storeD(t0, acc); lds_fence();
  // step 2: t1 = B * Q   (last use of tQ as staged data; t1 aliases tP)
  loadA_lds(tB, fa); loadB_lds(tQ, fb);
  acc = mm16(fa, fb, z); storeD(t1, acc); lds_fence();
  // step 3: Pp = (B Q) B^T + (A P) A^T   (B-operand of X^T == loadA_lds(X))
  loadA_lds(t1, fa); loadA_lds(tB, fb);
  acc = mm16(fa, fb, z);
  loadA_lds(t0, fa); loadA_lds(tA, fb);
  acc = mm16(fa, fb, acc);
  storeD(t0, acc); lds_fence();  // t0 = Pp (kept)
  // step 4: t1 = C * Pp
  loadA_lds(tC, fa); loadB_lds(t0, fb);
  acc = mm16(fa, fb, z); storeD(t1, acc); lds_fence();
  // step 5: S = (C Pp) C^T + R  -> t2 (aliases tQ; R tile zero-padded add)
  loadA_lds(t1, fa); loadA_lds(tC, fb);
  acc = mm16(fa, fb, z);
  {
    int c = lane & 15; int h = lane >> 4;
#pragma unroll
    for (int v = 0; v < 8; ++v) acc[v] += tR[(v + 8 * h) * LS + c];
  }
  storeD(t2, acc); lds_fence();

  // t1 = S^-1 (6x6 Gauss-Jordan across lanes 0..5; S is SPD -> no pivoting)
  storeD(t1, z); lds_fence();  // zero-pad Sinv tile
  {
    int j = lane < 6 ? lane : 0;
    float s[6], vi[6];
#pragma unroll
    for (int cc = 0; cc < 6; ++cc) {
      s[cc] = t2[j * LS + cc];
      vi[cc] = (cc == j) ? 1.0f : 0.0f;
    }
#pragma unroll
    for (int k = 0; k < 6; ++k) {
      float piv = __shfl(s[k], k, 32);
      float pinv = 1.0f / piv;
      float ps[6], pi[6];
#pragma unroll
      for (int cc = 0; cc < 6; ++cc) {
        ps[cc] = __shfl(s[cc], k, 32) * pinv;
        pi[cc] = __shfl(vi[cc], k, 32) * pinv;
      }
      if (lane == k) {
#pragma unroll
        for (int cc = 0; cc < 6; ++cc) { s[cc] = ps[cc]; vi[cc] = pi[cc]; }
      } else {
        float fval = s[k];
#pragma unroll
        for (int cc = 0; cc < 6; ++cc) {
          s[cc] -= fval * ps[cc];
          vi[cc] -= fval * pi[cc];
        }
      }
    }
    if (lane < 6) {
#pragma unroll
      for (int cc = 0; cc < 6; ++cc) t1[lane * LS + cc] = vi[cc];
    }
  }
  lds_fence();

  // t3 = Pp C^T
  loadA_lds(t0, fa); loadA_lds(tC, fb);
  acc = mm16(fa, fb, z); storeD(t3, acc); lds_fence();
  // t2 = K = (Pp C^T) Sinv
  loadA_lds(t3, fa); loadB_lds(t1, fb);
  acc = mm16(fa, fb, z); storeD(t2, acc); lds_fence();  // t2 = K (kept)
  // t3 = IKC = I - K C
  loadA_lds(t2, fa); loadB_lds(tC, fb);
  acc = mm16(fa, fb, z);
  {
    int c = lane & 15; int h = lane >> 4;
#pragma unroll
    for (int v = 0; v < 8; ++v) {
      int row = v + 8 * h;
      float idv = (row == c && row < 15) ? 1.0f : 0.0f;
      acc[v] = idv - acc[v];
    }
  }
  storeD(t3, acc); lds_fence();
  // t1 = IKC * Pp
  loadA_lds(t3, fa); loadB_lds(t0, fb);
  acc = mm16(fa, fb, z); storeD(t1, acc); lds_fence();
  // pn = (IKC Pp) IKC^T
  loadA_lds(t1, fa); loadA_lds(t3, fb);
  v8f pn = mm16(fa, fb, z);
  // t1 = K R
  loadA_lds(t2, fa); loadB_lds(tR, fb);
  acc = mm16(fa, fb, z); storeD(t1, acc); lds_fence();
  // pn += (K R) K^T
  loadA_lds(t1, fa); loadA_lds(t2, fb);
  pn = mm16(fa, fb, pn);

  // Park Pn in t3, then write it out with coalesced linear stores.
  storeD(t3, pn); lds_fence();
  {
#pragma unroll
    for (int base = 0; base < 225; base += 32) {
      int i = base + lane;
      if (i < 225) {
        int r = i / 15, c = i - r * 15;
        out_P[(size_t)b * 225 + i] = t3[r * LS + c];
      }
    }
  }

  // innovation e = obs - (C x + D u + c2)   (C from staged tile)
  if (lane < 6) {
    float a = c2b[lane];
#pragma unroll
    for (int t = 0; t < 15; ++t) a += tC[lane * LS + t] * xb[t];
#pragma unroll
    for (int t = 0; t < 6; ++t) a += Db[lane * 6 + t] * ub[t];
    ev[w][lane] = ob[lane] - a;
  }
  lds_fence();
  // xp = A x + B u + c1 + K e   (A, B, K from LDS tiles)
  if (lane < 15) {
    float a = c1b[lane];
#pragma unroll
    for (int t = 0; t < 15; ++t) a += tA[lane * LS + t] * xb[t];
#pragma unroll
    for (int t = 0; t < 6; ++t) a += tB[lane * LS + t] * ub[t];
#pragma unroll
    for (int t = 0; t < 6; ++t) a += t2[lane * LS + t] * ev[w][t];
    out_x[(size_t)b * 15 + lane] = a;
  }
}

extern "C" void kernel_launch(void* const* d_in, const int* in_sizes, int n_in,
                              void* d_out, int out_size, void* d_ws, size_t ws_size,
                              hipStream_t stream) {
  const float* state = (const float*)d_in[0];
  const float* obs   = (const float*)d_in[1];
  const float* u     = (const float*)d_in[2];
  const float* Pmat  = (const float*)d_in[3];
  const float* A     = (const float*)d_in[4];
  const float* Bm    = (const float*)d_in[5];
  const float* C     = (const float*)d_in[6];
  const float* D     = (const float*)d_in[7];
  const float* c1    = (const float*)d_in[8];
  const float* c2    = (const float*)d_in[9];
  const float* Q     = (const float*)d_in[10];
  const float* R     = (const float*)d_in[11];

  int BN = in_sizes[0] / 15;
  float* out_x = (float*)d_out;
  float* out_P = out_x + (size_t)BN * 15;

  int blocks = (BN + 3) / 4;  // 4 waves / block, 1 batch element / wave
  ekf_wmma_kernel<<<blocks, 128, 0, stream>>>(state, obs, u, Pmat, A, Bm, C, D,
                                              c1, c2, Q, R, out_x, out_P, BN);
}